// GCN_88132728914134
// MI455X (gfx1250) — compile-verified
//
#include <hip/hip_runtime.h>

// GCN on MI455X (gfx1250): dense h@W via V_WMMA_F32_16X16X4_F32 (fp32-exact),
// sparse scatter via global_atomic_add_f32.
// Working set (~77MB) fits L2-adjacent; GEMM stages A-slab + full W in LDS
// (76KB of the 320KB/WGP).

typedef __attribute__((ext_vector_type(2))) float v2f;
typedef __attribute__((ext_vector_type(8))) float v8f;

#define CH 128
#define TGT 10
#define NGRAPH 512

// ---------------- normalization ----------------
__global__ void k_deg_init(float* deg, int n) {
  int i = blockIdx.x * blockDim.x + threadIdx.x;
  if (i < n) deg[i] = 1.0f;  // self-loop contributes 1 to in-degree
}

__global__ void k_deg_accum(const int* __restrict__ dst, float* deg, int e) {
  int i = blockIdx.x * blockDim.x + threadIdx.x;
  if (i < e) atomicAdd(&deg[dst[i]], 1.0f);
}

__global__ void k_dinv(float* deg, int n) {
  int i = blockIdx.x * blockDim.x + threadIdx.x;
  if (i < n) deg[i] = rsqrtf(fmaxf(deg[i], 1.0f));
}

// ---------------- dense GEMM: out[N,128] = act(A)[N,128] @ W[128,128] -------
// Block: 256 threads = 8 waves. Block owns a 16-row slab; wave w owns cols
// [16w,16w+16). K=128 in 32 steps of V_WMMA_F32_16X16X4_F32.
// ISA layouts (7.12.2): A 16x4 f32 -> lane L<16: M=L,K={0,1}; L>=16: K={2,3}.
// B 4x16 mirrors C's half-split: VGPR0 = rows {0,2}, VGPR1 = rows {1,3}.
// D: VGPR r -> row r (lanes 0-15) / row r+8 (lanes 16-31), col = lane%16.
__global__ __launch_bounds__(256) void k_gemm_wmma(
    const float* __restrict__ A, const float* __restrict__ W,
    float* __restrict__ out, int n, int relu) {
  __shared__ float sA[16 * 132];    // padded: conflict-free A reads
  __shared__ float sB[128 * 132];
  const int tid  = threadIdx.x;
  const int lane = tid & 31;
  const int wave = tid >> 5;
  const int row0 = blockIdx.x * 16;

  for (int i = tid; i < 16 * CH; i += 256) {
    int r = i >> 7, c = i & (CH - 1);
    float v = (row0 + r < n) ? A[(row0 + r) * CH + c] : 0.0f;
    if (relu) v = fmaxf(v, 0.0f);
    sA[r * 132 + c] = v;
  }
  for (int i = tid; i < CH * CH; i += 256) {
    int r = i >> 7, c = i & (CH - 1);
    sB[r * 132 + c] = W[i];
  }
  __syncthreads();

  const int col0  = wave << 4;
  const int m     = lane & 15;
  const int khalf = (lane >> 4) << 1;  // 0 or 2
  v8f acc = {};
#pragma unroll
  for (int k0 = 0; k0 < CH; k0 += 4) {
    v2f a, b;
    a.x = sA[m * 132 + k0 + khalf];
    a.y = sA[m * 132 + k0 + khalf + 1];
    b.x = sB[(k0 + khalf) * 132 + col0 + m];
    b.y = sB[(k0 + khalf + 1) * 132 + col0 + m];
    acc = __builtin_amdgcn_wmma_f32_16x16x4_f32(false, a, false, b,
                                                (short)0, acc, false, false);
  }

  const int nCol  = col0 + m;
  const int mbase = (lane >> 4) << 3;  // 0 or 8
#pragma unroll
  for (int r = 0; r < 8; ++r) {
    int row = row0 + mbase + r;
    if (row < n) out[row * CH + nCol] = acc[r];
  }
}

// -------- aggregation: out = dinv[i]^2 * tmp[i] + b  (self-loop + bias) -----
__global__ void k_agg_init(const float* __restrict__ tmp,
                           const float* __restrict__ dinv,
                           const float* __restrict__ bias,
                           float* __restrict__ out, int n) {
  int i = blockIdx.x * blockDim.x + threadIdx.x;
  if (i < n * CH) {
    int node = i >> 7;
    int c    = i & (CH - 1);
    float di = dinv[node];
    out[i] = di * di * tmp[i] + bias[c];
  }
}

// one wave (32 lanes) per edge; each lane scatters 4 channels
__global__ void k_edge_agg(const float* __restrict__ tmp,
                           const int* __restrict__ src,
                           const int* __restrict__ dst,
                           const float* __restrict__ dinv,
                           float* __restrict__ out, int e) {
  int t = blockIdx.x * blockDim.x + threadIdx.x;
  if (t >= e * 32) return;
  int edge = t >> 5;
  int c    = (t & 31) << 2;
  int s = src[edge], d = dst[edge];
  float w = dinv[s] * dinv[d];
  const float4 v = *(const float4*)(tmp + (size_t)s * CH + c);
  float* o = out + (size_t)d * CH + c;
  atomicAdd(o + 0, w * v.x);
  atomicAdd(o + 1, w * v.y);
  atomicAdd(o + 2, w * v.z);
  atomicAdd(o + 3, w * v.w);
}

// ---------------- pooling + head ----------------
__global__ void k_zero(float* p, int n) {
  int i = blockIdx.x * blockDim.x + threadIdx.x;
  if (i < n) p[i] = 0.0f;
}

__global__ void k_pool_accum(const float* __restrict__ h,
                             const int* __restrict__ batch,
                             float* __restrict__ psum,
                             float* __restrict__ pcnt, int n) {
  int t = blockIdx.x * blockDim.x + threadIdx.x;
  if (t >= n * 32) return;
  int node = t >> 5;
  int c    = (t & 31) << 2;
  int g = batch[node];
  const float4 v = *(const float4*)(h + (size_t)node * CH + c);
  float* o = psum + (size_t)g * CH + c;
  atomicAdd(o + 0, v.x);
  atomicAdd(o + 1, v.y);
  atomicAdd(o + 2, v.z);
  atomicAdd(o + 3, v.w);
  if ((t & 31) == 0) atomicAdd(&pcnt[g], 1.0f);
}

__global__ void k_head(const float* __restrict__ psum,
                       const float* __restrict__ pcnt,
                       const float* __restrict__ Wlin,
                       const float* __restrict__ blin,
                       float* __restrict__ out) {
  int i = blockIdx.x * blockDim.x + threadIdx.x;
  if (i >= NGRAPH * TGT) return;
  int g = i / TGT;
  int t = i % TGT;
  float inv = 1.0f / fmaxf(pcnt[g], 1.0f);
  float acc = blin[t];
#pragma unroll 4
  for (int c = 0; c < CH; ++c)
    acc += psum[g * CH + c] * inv * Wlin[c * TGT + t];
  out[i] = acc;
}

// ---------------- launcher ----------------
extern "C" void kernel_launch(void* const* d_in, const int* in_sizes, int n_in,
                              void* d_out, int out_size, void* d_ws,
                              size_t ws_size, hipStream_t stream) {
  const float* x     = (const float*)d_in[0];
  const int*   ei    = (const int*)d_in[1];
  const int*   batch = (const int*)d_in[2];
  const float* W1    = (const float*)d_in[3];
  const float* b1    = (const float*)d_in[4];
  const float* Ws    = (const float*)d_in[5];
  const float* bs    = (const float*)d_in[6];
  const float* Wlin  = (const float*)d_in[7];
  const float* blin  = (const float*)d_in[8];
  float* out = (float*)d_out;

  const int N = in_sizes[0] / CH;
  const int E = in_sizes[1] / 2;
  const int* src = ei;
  const int* dst = ei + E;

  // workspace partition (256B aligned)
  char* p = (char*)d_ws;
  auto take = [&p](size_t bytes) {
    void* r = (void*)p;
    p += (bytes + 255) & ~(size_t)255;
    return r;
  };
  float* dinv = (float*)take((size_t)N * 4);
  float* tmp  = (float*)take((size_t)N * CH * 4);
  float* h0   = (float*)take((size_t)N * CH * 4);
  float* h1   = (float*)take((size_t)N * CH * 4);
  float* psum = (float*)take((size_t)NGRAPH * CH * 4);
  float* pcnt = (float*)take((size_t)NGRAPH * 4);

  const int TB = 256;
  auto blks = [](long long n, int tb) { return (unsigned)((n + tb - 1) / tb); };

  // 1) symmetric normalization coefficients
  k_deg_init<<<blks(N, TB), TB, 0, stream>>>(dinv, N);
  k_deg_accum<<<blks(E, TB), TB, 0, stream>>>(dst, dinv, E);
  k_dinv<<<blks(N, TB), TB, 0, stream>>>(dinv, N);

  const unsigned gGemm = blks(N, 16);
  const unsigned gNC   = blks((long long)N * CH, TB);
  const unsigned gEdge = blks((long long)E * 32, TB);

  // 2) layer 1: x @ W1 -> agg (+relu folded into next gemm's load)
  k_gemm_wmma<<<gGemm, TB, 0, stream>>>(x, W1, tmp, N, 0);
  k_agg_init<<<gNC, TB, 0, stream>>>(tmp, dinv, b1, h0, N);
  k_edge_agg<<<gEdge, TB, 0, stream>>>(tmp, src, dst, dinv, h0, E);

  // 3) layers 2..4 (Ws[0..2]); relu on input for all (post-L1/L2/L3 relu)
  k_gemm_wmma<<<gGemm, TB, 0, stream>>>(h0, Ws + 0 * CH * CH, tmp, N, 1);
  k_agg_init<<<gNC, TB, 0, stream>>>(tmp, dinv, bs + 0 * CH, h1, N);
  k_edge_agg<<<gEdge, TB, 0, stream>>>(tmp, src, dst, dinv, h1, E);

  k_gemm_wmma<<<gGemm, TB, 0, stream>>>(h1, Ws + 1 * CH * CH, tmp, N, 1);
  k_agg_init<<<gNC, TB, 0, stream>>>(tmp, dinv, bs + 1 * CH, h0, N);
  k_edge_agg<<<gEdge, TB, 0, stream>>>(tmp, src, dst, dinv, h0, E);

  k_gemm_wmma<<<gGemm, TB, 0, stream>>>(h0, Ws + 2 * CH * CH, tmp, N, 1);
  k_agg_init<<<gNC, TB, 0, stream>>>(tmp, dinv, bs + 2 * CH, h1, N);
  k_edge_agg<<<gEdge, TB, 0, stream>>>(tmp, src, dst, dinv, h1, E);

  // 4) mean pool + linear head
  k_zero<<<blks(NGRAPH * CH, TB), TB, 0, stream>>>(psum, NGRAPH * CH);
  k_zero<<<blks(NGRAPH, TB), TB, 0, stream>>>(pcnt, NGRAPH);
  k_pool_accum<<<blks((long long)N * 32, TB), TB, 0, stream>>>(h1, batch, psum,
                                                               pcnt, N);
  k_head<<<blks(NGRAPH * TGT, TB), TB, 0, stream>>>(psum, pcnt, Wlin, blin,
                                                    out);
}